// Rx_layer_47871705481511
// MI455X (gfx1250) — compile-verified
//
#include <hip/hip_runtime.h>
#include <stdint.h>

typedef float v4f __attribute__((ext_vector_type(4)));

#define TILE 256  // qubits per block-tile == blockDim.x

// Wave-relative LDS byte offset of a __shared__ object: the flat->LDS mapping
// uses addr[31:0], which equals the AS(3) offset the async engine expects.
__device__ __forceinline__ uint32_t lds_off(const void* p) {
    return (uint32_t)(uintptr_t)p;
}

// out[i][k] = c*S[i][k] - 1j*s*S[1-i][k]
__device__ __forceinline__ void rx_apply(float wq, v4f r, v4f i, v4f& o0, v4f& o1) {
    float s, c;
    __sincosf(0.5f * wq, &s, &c);            // v_sin_f32 / v_cos_f32 fast path
    o0.x = __builtin_fmaf(c, r.x,  s * i.z); // re(0,0)
    o0.y = __builtin_fmaf(c, i.x, -s * r.z); // im(0,0)
    o0.z = __builtin_fmaf(c, r.y,  s * i.w); // re(0,1)
    o0.w = __builtin_fmaf(c, i.y, -s * r.w); // im(0,1)
    o1.x = __builtin_fmaf(c, r.z,  s * i.x); // re(1,0)
    o1.y = __builtin_fmaf(c, i.z, -s * r.x); // im(1,0)
    o1.z = __builtin_fmaf(c, r.w,  s * i.y); // re(1,1)
    o1.w = __builtin_fmaf(c, i.w, -s * r.y); // im(1,1)
}

__global__ __launch_bounds__(TILE) void rx_layer_async(
    const float* __restrict__ w,
    const v4f*  __restrict__ re,
    const v4f*  __restrict__ im,
    v4f*        __restrict__ out,
    int n)
{
    __shared__ float s_w [2][TILE];
    __shared__ v4f   s_re[2][TILE];
    __shared__ v4f   s_im[2][TILE];

    const int tid    = threadIdx.x;
    const int ntiles = n / TILE;

    // Issue one tile's async global->LDS copies into buffer b.
    // Each lane copies exactly the slots it will later read: dependency is
    // wave-local, tracked purely by ASYNCcnt (no workgroup barrier needed).
    auto issue = [&](int tile, int b) {
        const uint32_t q     = (uint32_t)tile * TILE + (uint32_t)tid;
        const uint32_t off4  = q * 4u;     // byte offset into w
        const uint32_t off16 = q * 16u;    // byte offset into re/im
        const uint32_t lw    = lds_off(&s_w [b][tid]);
        const uint32_t lre   = lds_off(&s_re[b][tid]);
        const uint32_t lim   = lds_off(&s_im[b][tid]);
        asm volatile("global_load_async_to_lds_b32 %0, %1, %2"
                     :: "v"(lw),  "v"(off4),  "s"(w)  : "memory");
        asm volatile("global_load_async_to_lds_b128 %0, %1, %2"
                     :: "v"(lre), "v"(off16), "s"(re) : "memory");
        asm volatile("global_load_async_to_lds_b128 %0, %1, %2"
                     :: "v"(lim), "v"(off16), "s"(im) : "memory");
    };

    // ---- double-buffered async pipeline over full tiles ----
    int t = blockIdx.x;
    if (t < ntiles) issue(t, 0);
    int buf = 0;
    for (; t < ntiles; t += gridDim.x, buf ^= 1) {
        const int tn = t + gridDim.x;
        if (tn < ntiles) {
            issue(tn, buf ^ 1);
            // wait for tile t's 3 copies; leave tile tn's 3 in flight
            asm volatile("s_wait_asynccnt 0x3" ::: "memory");
        } else {
            asm volatile("s_wait_asynccnt 0x0" ::: "memory");
        }

        const float wq = s_w [buf][tid];   // ds_load_b32
        const v4f   r  = s_re[buf][tid];   // ds_load_b128
        const v4f   i  = s_im[buf][tid];   // ds_load_b128

        v4f o0, o1;
        rx_apply(wq, r, i, o0, o1);

        const int q = t * TILE + tid;
        __builtin_nontemporal_store(o0, &out[2 * q]);     // b128 NT
        __builtin_nontemporal_store(o1, &out[2 * q + 1]); // b128 NT
    }

    // ---- tail (n not a multiple of TILE; empty for L = 4M) ----
    for (int q = ntiles * TILE + blockIdx.x * TILE + tid; q < n;
         q += (int)(gridDim.x * TILE)) {
        const float wq = __builtin_nontemporal_load(&w[q]);
        const v4f   r  = __builtin_nontemporal_load(&re[q]);
        const v4f   i  = __builtin_nontemporal_load(&im[q]);
        v4f o0, o1;
        rx_apply(wq, r, i, o0, o1);
        __builtin_nontemporal_store(o0, &out[2 * q]);
        __builtin_nontemporal_store(o1, &out[2 * q + 1]);
    }
}

extern "C" void kernel_launch(void* const* d_in, const int* in_sizes, int n_in,
                              void* d_out, int out_size, void* d_ws, size_t ws_size,
                              hipStream_t stream) {
    const float* w  = (const float*)d_in[0];   // weights (L,)
    const v4f*   re = (const v4f*)  d_in[1];   // state_re (L,2,2)
    const v4f*   im = (const v4f*)  d_in[2];   // state_im (L,2,2)
    v4f*         out = (v4f*)d_out;            // complex64 (L,2,2) as float pairs

    const int n = in_sizes[0];                 // L = 4194304
    int blocks = (n + TILE - 1) / TILE;
    if (blocks > 4096) blocks = 4096;          // 4 tiles per block at L=4M;
                                               // 32768 wave32 waves in flight
    rx_layer_async<<<blocks, TILE, 0, stream>>>(w, re, im, out, n);
}